// Model_38225208935040
// MI455X (gfx1250) — compile-verified
//
#include <hip/hip_runtime.h>
#include <math.h>

// v2f: A/B fragment of V_WMMA_F32_16X16X4_F32 (16x4 / 4x16 f32, 2 VGPRs/lane)
// v8f: C/D fragment (16x16 f32, 8 VGPRs/lane)
typedef float v2f __attribute__((ext_vector_type(2)));
typedef float v8f __attribute__((ext_vector_type(8)));

static __device__ __forceinline__ v8f wmma4(v2f a, v2f b, v8f c) {
    // (neg_a, A, neg_b, B, c_mod, C, reuse_a, reuse_b)
    return __builtin_amdgcn_wmma_f32_16x16x4_f32(false, a, false, b, (short)0, c,
                                                 false, false);
}

// Single range-reduction sin+cos (OCML sincos) — halves trig VALU cost vs
// separate sinf/cosf calls, which each expand the Payne-Hanek big-arg path.
// HIP signature: void sincosf(float, float* sin, float* cos).
static __device__ __forceinline__ void sc(float z, float& s, float& c) {
    float ss, cc;
    sincosf(z, &ss, &cc);
    s = ss;
    c = cc;
}

// ---------------------------------------------------------------------------
// Precompute per-region activation-mean tables:
//   A_r[k] = wa*C_r[k] - wb*S_r[k],  B_r[k] = wa*S_r[k] + wb*C_r[k]
// where C/S = mean over the perturbation grid of cos/sin(p . W1[:,k]).
// Region 0 (unperturbed): A=wa, B=wb.
// tab layout: tab[0..255] = A[r][k], tab[256..511] = B[r][k]
// ---------------------------------------------------------------------------
__global__ __launch_bounds__(64) void precompute_regions(
    const float* __restrict__ emb_W1, const float* __restrict__ emb_wa,
    const float* __restrict__ emb_wb, float* __restrict__ tab) {
    const int k = threadIdx.x;  // 0..63 (hidden unit)
    const float wa = emb_wa[0], wb = emb_wb[0];
    tab[k] = wa;
    tab[256 + k] = wb;
    const float w0 = emb_W1[k], w1 = emb_W1[64 + k], w2 = emb_W1[128 + k];
    const float ranges[3] = {0.01f, 0.05f, 0.09f};
    const int sizes[3] = {3, 5, 7};
    for (int r = 0; r < 3; ++r) {
        const int s = sizes[r];
        const float rr = ranges[r];
        const float step = 2.0f * rr / (float)(s - 1);
        float sumC = 0.0f, sumS = 0.0f;
        for (int i = 0; i < s; ++i) {
            const float px = -rr + step * (float)i;
            for (int j = 0; j < s; ++j) {
                const float py = -rr + step * (float)j;
                const float base = px * w0 + py * w1;
                for (int l = 0; l < s; ++l) {
                    const float pt = -rr + step * (float)l;
                    float sv, cv;
                    sc(base + pt * w2, sv, cv);
                    sumS += sv;
                    sumC += cv;
                }
            }
        }
        const float inv = 1.0f / (float)(s * s * s);
        const float C = sumC * inv, S = sumS * inv;
        tab[(r + 1) * 64 + k] = wa * C - wb * S;
        tab[256 + (r + 1) * 64 + k] = wa * S + wb * C;
    }
}

// ---------------------------------------------------------------------------
// Main kernel: one wave32 per block, each wave owns a 16-sample tile.
// ---------------------------------------------------------------------------
__global__ __launch_bounds__(32) void pinn_forward(
    const float* __restrict__ x, const float* __restrict__ y,
    const float* __restrict__ t,
    const float* __restrict__ emb_W1, const float* __restrict__ emb_b1,
    const float* __restrict__ emb_W2, const float* __restrict__ emb_b2,
    const float* __restrict__ mix_W1, const float* __restrict__ mix_b1,
    const float* __restrict__ mix_wa, const float* __restrict__ mix_wb,
    const float* __restrict__ mix_W2, const float* __restrict__ mix_b2,
    const float* __restrict__ out_W0, const float* __restrict__ out_b0,
    const float* __restrict__ out_wa0, const float* __restrict__ out_wb0,
    const float* __restrict__ out_Wh, const float* __restrict__ out_bh,
    const float* __restrict__ out_wah, const float* __restrict__ out_wbh,
    const float* __restrict__ out_Wf1, const float* __restrict__ out_bf1,
    const float* __restrict__ out_waf, const float* __restrict__ out_wbf,
    const float* __restrict__ out_Wf2, const float* __restrict__ out_bf2,
    const float* __restrict__ tab, float* __restrict__ out) {
    __shared__ float s_act[4][16][65];  // [region][sample][unit], padded
    __shared__ float s_e[16][257];      // [sample][channel], padded (odd stride)
    __shared__ float s_h[2][16][65];    // ping-pong hidden [sample][unit]

    const int lane = threadIdx.x & 31;
    const int m = lane & 15;  // WMMA row (A) / col (B,C)
    const int hi = lane >> 4; // half-wave selector
    const int base = blockIdx.x * 16;

    // ---- Stage 1: hb = [x,y,t,1] @ [W1; b1]  (4 WMMAs, K=4 incl. bias col)
    //      then act_r = sin(hb)*A_r + cos(hb)*B_r  -> LDS
    v2f a;
    if (hi == 0) { a.x = x[base + m]; a.y = y[base + m]; }
    else         { a.x = t[base + m]; a.y = 1.0f; }
    for (int t4 = 0; t4 < 4; ++t4) {
        const int u = t4 * 16 + m;
        v2f b;
        if (hi == 0) { b.x = emb_W1[u];       b.y = emb_W1[64 + u]; }
        else         { b.x = emb_W1[128 + u]; b.y = emb_b1[u]; }
        v8f c = {};
        c = wmma4(a, b, c);
        const float A0 = tab[u],       B0 = tab[256 + u];
        const float A1 = tab[64 + u],  B1 = tab[320 + u];
        const float A2 = tab[128 + u], B2 = tab[384 + u];
        const float A3 = tab[192 + u], B3 = tab[448 + u];
#pragma unroll
        for (int i = 0; i < 8; ++i) {
            const int sm = i + 8 * hi;
            float sh, ch;
            sc(c[i], sh, ch);
            s_act[0][sm][u] = sh * A0 + ch * B0;
            s_act[1][sm][u] = sh * A1 + ch * B1;
            s_act[2][sm][u] = sh * A2 + ch * B2;
            s_act[3][sm][u] = sh * A3 + ch * B3;
        }
    }
    __syncthreads();

    // Mixer parameters into registers (fully unrolled use keeps them in VGPRs)
    float mW1r[32], mb1r[8], mW2r[8];
#pragma unroll
    for (int q = 0; q < 32; ++q) mW1r[q] = mix_W1[q];
#pragma unroll
    for (int q = 0; q < 8; ++q) { mb1r[q] = mix_b1[q]; mW2r[q] = mix_W2[q]; }
    const float mwa = mix_wa[0], mwb = mix_wb[0], mb2 = mix_b2[0];

    // ---- Stage 2: per 16-col tile of the 256 channels:
    //      feats_r = act_r @ W2 + b2 (4 regions, K=64), then elementwise
    //      4->8->1 wave-mixer -> e[sample][channel] in LDS
    for (int ct = 0; ct < 16; ++ct) {
        const int n = ct * 16 + m;
        v8f f0 = {}, f1 = {}, f2 = {}, f3 = {};
#pragma unroll
        for (int kk = 0; kk < 16; ++kk) {
            const int k0 = kk * 4 + 2 * hi;
            v2f bf;
            bf.x = emb_W2[k0 * 256 + n];
            bf.y = emb_W2[(k0 + 1) * 256 + n];
            v2f af;
            af.x = s_act[0][m][k0]; af.y = s_act[0][m][k0 + 1];
            f0 = wmma4(af, bf, f0);
            af.x = s_act[1][m][k0]; af.y = s_act[1][m][k0 + 1];
            f1 = wmma4(af, bf, f1);
            af.x = s_act[2][m][k0]; af.y = s_act[2][m][k0 + 1];
            f2 = wmma4(af, bf, f2);
            af.x = s_act[3][m][k0]; af.y = s_act[3][m][k0 + 1];
            f3 = wmma4(af, bf, f3);
        }
        const float b2n = emb_b2[n];
#pragma unroll
        for (int i = 0; i < 8; ++i) {
            const int sm = i + 8 * hi;
            const float g0 = f0[i] + b2n, g1 = f1[i] + b2n;
            const float g2 = f2[i] + b2n, g3 = f3[i] + b2n;
            float acc = 0.0f;
#pragma unroll
            for (int j = 0; j < 8; ++j) {
                const float z = g0 * mW1r[j] + g1 * mW1r[8 + j] +
                                g2 * mW1r[16 + j] + g3 * mW1r[24 + j] + mb1r[j];
                float sz, cz;
                sc(z, sz, cz);
                acc += (mwa * sz + mwb * cz) * mW2r[j];
            }
            s_e[sm][n] = acc + mb2;
        }
    }
    __syncthreads();

    // ---- Stage 3: h = wave(e @ out_W0 + b0)  (K=256 -> 64 k-steps, 4 col tiles)
    {
        const float wa0 = out_wa0[0], wb0 = out_wb0[0];
        for (int ot = 0; ot < 4; ++ot) {
            const int n = ot * 16 + m;
            v8f c = {};
            for (int kk = 0; kk < 64; ++kk) {
                const int k0 = kk * 4 + 2 * hi;
                v2f af; af.x = s_e[m][k0]; af.y = s_e[m][k0 + 1];
                v2f bf;
                bf.x = out_W0[k0 * 64 + n];
                bf.y = out_W0[(k0 + 1) * 64 + n];
                c = wmma4(af, bf, c);
            }
            const float bn = out_b0[n];
#pragma unroll
            for (int i = 0; i < 8; ++i) {
                float sz, cz;
                sc(c[i] + bn, sz, cz);
                s_h[0][i + 8 * hi][n] = wa0 * sz + wb0 * cz;
            }
        }
    }
    __syncthreads();

    // ---- Stage 4: 3 hidden layers + f1, all 64x64 (K=64 -> 16 k-steps)
    for (int l = 0; l < 4; ++l) {
        const float* W = (l < 3) ? (out_Wh + l * 64 * 64) : out_Wf1;
        const float* bb = (l < 3) ? (out_bh + l * 64) : out_bf1;
        const float wa = (l < 3) ? out_wah[l] : out_waf[0];
        const float wb = (l < 3) ? out_wbh[l] : out_wbf[0];
        const int src = l & 1, dst = src ^ 1;
        for (int ot = 0; ot < 4; ++ot) {
            const int n = ot * 16 + m;
            v8f c = {};
#pragma unroll
            for (int kk = 0; kk < 16; ++kk) {
                const int k0 = kk * 4 + 2 * hi;
                v2f af; af.x = s_h[src][m][k0]; af.y = s_h[src][m][k0 + 1];
                v2f bf;
                bf.x = W[k0 * 64 + n];
                bf.y = W[(k0 + 1) * 64 + n];
                c = wmma4(af, bf, c);
            }
            const float bn = bb[n];
#pragma unroll
            for (int i = 0; i < 8; ++i) {
                float sz, cz;
                sc(c[i] + bn, sz, cz);
                s_h[dst][i + 8 * hi][n] = wa * sz + wb * cz;
            }
        }
        __syncthreads();
    }

    // ---- Stage 5: out = h @ Wf2 + bf2  ([16x64]x[64x3], scalar dot products)
    // After l=3 the result lives in s_h[0].
    for (int idx = lane; idx < 48; idx += 32) {
        const int sm = idx / 3, o = idx % 3;
        float acc = out_bf2[o];
#pragma unroll
        for (int k = 0; k < 64; ++k) acc += s_h[0][sm][k] * out_Wf2[k * 3 + o];
        out[(base + sm) * 3 + o] = acc;
    }
}

extern "C" void kernel_launch(void* const* d_in, const int* in_sizes, int n_in,
                              void* d_out, int out_size, void* d_ws, size_t ws_size,
                              hipStream_t stream) {
    const float* x      = (const float*)d_in[0];
    const float* y      = (const float*)d_in[1];
    const float* t      = (const float*)d_in[2];
    const float* emb_W1 = (const float*)d_in[3];
    const float* emb_b1 = (const float*)d_in[4];
    const float* emb_wa = (const float*)d_in[5];
    const float* emb_wb = (const float*)d_in[6];
    const float* emb_W2 = (const float*)d_in[7];
    const float* emb_b2 = (const float*)d_in[8];
    const float* mix_W1 = (const float*)d_in[9];
    const float* mix_b1 = (const float*)d_in[10];
    const float* mix_wa = (const float*)d_in[11];
    const float* mix_wb = (const float*)d_in[12];
    const float* mix_W2 = (const float*)d_in[13];
    const float* mix_b2 = (const float*)d_in[14];
    const float* out_W0 = (const float*)d_in[15];
    const float* out_b0 = (const float*)d_in[16];
    const float* out_wa0 = (const float*)d_in[17];
    const float* out_wb0 = (const float*)d_in[18];
    const float* out_Wh  = (const float*)d_in[19];
    const float* out_bh  = (const float*)d_in[20];
    const float* out_wah = (const float*)d_in[21];
    const float* out_wbh = (const float*)d_in[22];
    const float* out_Wf1 = (const float*)d_in[23];
    const float* out_bf1 = (const float*)d_in[24];
    const float* out_waf = (const float*)d_in[25];
    const float* out_wbf = (const float*)d_in[26];
    const float* out_Wf2 = (const float*)d_in[27];
    const float* out_bf2 = (const float*)d_in[28];

    float* tab = (float*)d_ws;  // 512 floats: A[4][64], B[4][64]

    precompute_regions<<<1, 64, 0, stream>>>(emb_W1, emb_wa, emb_wb, tab);
    pinn_forward<<<256, 32, 0, stream>>>(
        x, y, t, emb_W1, emb_b1, emb_W2, emb_b2,
        mix_W1, mix_b1, mix_wa, mix_wb, mix_W2, mix_b2,
        out_W0, out_b0, out_wa0, out_wb0, out_Wh, out_bh, out_wah, out_wbh,
        out_Wf1, out_bf1, out_waf, out_wbf, out_Wf2, out_bf2,
        tab, (float*)d_out);
}